// CTCLossWrapper_31035433681490
// MI455X (gfx1250) — compile-verified
//
#include <hip/hip_runtime.h>
#include <cstdint>

#define B_ 64
#define T_ 1024
#define K_ 512
#define L_ 100
#define S_ 201     // 2*L+1
#define NSLOT 9    // LDS ring slots (PF+1 so a slot is never overwritten while readable)
#define PF 8       // prefetch depth in time steps (7 async ops/step -> 56 <= ASYNCcnt max 63)

// ---------------------------------------------------------------------------
// Kernel 1: per-(b,t) logsumexp over K=512.  One wave32 per row, 8 rows/block.
// Streams the full 128 MiB of logits (the bandwidth floor) and leaves it in L2.
// ---------------------------------------------------------------------------
__global__ __launch_bounds__(256) void lse_kernel(const float* __restrict__ logits,
                                                  float* __restrict__ lse) {
  const int lane = threadIdx.x & 31;
  const int w    = threadIdx.x >> 5;
  const size_t row = (size_t)blockIdx.x * 8 + w;          // < 65536
  const float* p = logits + row * K_;
  float4 v[4];
#pragma unroll
  for (int k = 0; k < 4; ++k)                             // lane-contiguous 512B per load
    v[k] = *(const float4*)(p + k * 128 + lane * 4);
  float m = v[0].x;
#pragma unroll
  for (int k = 0; k < 4; ++k)
    m = fmaxf(m, fmaxf(fmaxf(v[k].x, v[k].y), fmaxf(v[k].z, v[k].w)));
#pragma unroll
  for (int off = 16; off > 0; off >>= 1) m = fmaxf(m, __shfl_xor(m, off, 32));
  float s = 0.f;
#pragma unroll
  for (int k = 0; k < 4; ++k)
    s += __expf(v[k].x - m) + __expf(v[k].y - m) + __expf(v[k].z - m) + __expf(v[k].w - m);
#pragma unroll
  for (int off = 16; off > 0; off >>= 1) s += __shfl_xor(s, off, 32);
  if (lane == 0) lse[row] = m + logf(s);
}

// ---------------------------------------------------------------------------
// Kernel 2: scaled CTC forward recurrence.  One wave32 per batch element.
// Lane l owns states s = 7l .. 7l+6 (lanes 0..28 cover S=201; rest masked).
// Gathers for step t+8 are issued with global_load_async_to_lds_b32 into a
// 9-slot LDS ring and drained with s_wait_asynccnt (CDNA5 async path).
// ---------------------------------------------------------------------------
__global__ __launch_bounds__(32) void ctc_forward_kernel(const float* __restrict__ logits,
                                                         const int* __restrict__ targets,
                                                         const float* __restrict__ lse,
                                                         float* __restrict__ perBatch) {
  __shared__ float ring[NSLOT][224];                      // 9 * 896 B = 8064 B LDS
  const int lane = threadIdx.x;
  const int b    = blockIdx.x;
  const int* tg  = targets + b * L_;

  int   cols[7];
  float skipf[7], vmask[7], inj[7], a[7], pexp[7];
#pragma unroll
  for (int i = 0; i < 7; ++i) {
    const int  s     = lane * 7 + i;
    const bool valid = (s < S_);
    int col = 0; float sk = 0.f;
    if (valid && (s & 1)) {                               // odd s -> label position j
      const int j = (s - 1) >> 1;                         // 0..99, in range
      col = tg[j];
      if (j >= 1 && col != tg[j - 1]) sk = 1.f;           // skip transition allowed
    }
    cols[i]  = col;
    skipf[i] = sk;
    vmask[i] = valid ? 1.f : 0.f;
    inj[i]   = (lane == 0 && i < 2) ? 1.f : 0.f;          // alpha0 injection at t==0
    a[i]     = 0.f;
  }

  // Deterministic per-batch sum of lse over T (fixed-order tree).
  float sl = 0.f;
#pragma unroll
  for (int k = 0; k < 32; ++k) sl += lse[b * T_ + lane + 32 * k];
#pragma unroll
  for (int off = 16; off > 0; off >>= 1) sl += __shfl_xor(sl, off, 32);

  // Low 32 bits of a flat pointer into LDS == wave-relative LDS byte offset.
  const uint32_t lds0 = (uint32_t)(uintptr_t)(&ring[0][0]) + (uint32_t)lane * 28u;

  // Prologue: async-prefetch gathered rows for t = 0..PF-1 into slots 0..PF-1.
#pragma unroll
  for (int tt = 0; tt < PF; ++tt) {
    const float*   rowp = logits + ((size_t)b * T_ + tt) * K_;
    const uint32_t ldsb = lds0 + (uint32_t)tt * (224u * 4u);
#pragma unroll
    for (int i = 0; i < 7; ++i) {
      const uint64_t ga = (uint64_t)(uintptr_t)(rowp + cols[i]);
      asm volatile("global_load_async_to_lds_b32 %0, %1, off"
                   :: "v"(ldsb + 4u * i), "v"(ga) : "memory");
    }
  }
  asm volatile("s_wait_asynccnt 49" ::: "memory");        // slot 0 complete (in-order)
#pragma unroll
  for (int i = 0; i < 7; ++i) pexp[i] = vmask[i] * __expf(ring[0][lane * 7 + i]);

  double logC = 0.0;
  for (int t0 = 0; t0 < T_; t0 += 4) {
#pragma unroll
    for (int u = 0; u < 4; ++u) {
      const int t = t0 + u;
      // 56 in flight at loop top; <=42 remaining guarantees slots t and t+1 done.
      asm volatile("s_wait_asynccnt 42" ::: "memory");

      // Stage next step's raw gathered logits out of LDS (off the alpha chain).
      const int tn = (t + 1 < T_) ? (t + 1) : (T_ - 1);
      const int sn = tn % NSLOT;
      float praw[7];
#pragma unroll
      for (int i = 0; i < 7; ++i) praw[i] = ring[sn][lane * 7 + i];

      // Issue async gathers for step t+PF (clamped at the tail; never read then).
      {
        const int tp = (t + PF < T_) ? (t + PF) : (T_ - 1);
        const int sp = (t + PF) % NSLOT;
        const float*   rowp = logits + ((size_t)b * T_ + tp) * K_;
        const uint32_t ldsb = lds0 + (uint32_t)sp * (224u * 4u);
#pragma unroll
        for (int i = 0; i < 7; ++i) {
          const uint64_t ga = (uint64_t)(uintptr_t)(rowp + cols[i]);
          asm volatile("global_load_async_to_lds_b32 %0, %1, off"
                       :: "v"(ldsb + 4u * i), "v"(ga) : "memory");
        }
      }

      // --- CTC step t: alpha' = (a[s] + a[s-1] + skip*a[s-2] + inj@t0) * p ---
      const float injs = (t == 0) ? 1.f : 0.f;
      float hi6 = __shfl_up(a[6], 1, 32);                 // neighbor a[s-1] boundary
      float hi5 = __shfl_up(a[5], 1, 32);                 // neighbor a[s-2] boundary
      if (lane == 0) { hi5 = 0.f; hi6 = 0.f; }
      float na[7];
      na[0] = (a[0] + hi6  + skipf[0] * hi5  + inj[0] * injs) * pexp[0];
      na[1] = (a[1] + a[0] + skipf[1] * hi6  + inj[1] * injs) * pexp[1];
#pragma unroll
      for (int i = 2; i < 7; ++i)
        na[i] = (a[i] + a[i - 1] + skipf[i] * a[i - 2] + inj[i] * injs) * pexp[i];
#pragma unroll
      for (int i = 0; i < 7; ++i) a[i] = na[i];

      // exp for the NEXT step — independent of the alpha dependence chain.
#pragma unroll
      for (int i = 0; i < 7; ++i) pexp[i] = vmask[i] * __expf(praw[i]);

      // Renormalize every 4 steps; accumulate log of the scale.
      if (u == 3) {
        float c = ((a[0] + a[1]) + (a[2] + a[3])) + ((a[4] + a[5]) + a[6]);
#pragma unroll
        for (int off = 16; off > 0; off >>= 1) c += __shfl_xor(c, off, 32);
        if (c > 0.f) {
          const float inv = 1.f / c;
#pragma unroll
          for (int i = 0; i < 7; ++i) a[i] *= inv;
          logC += (double)logf(c);
        }
      }
    }
  }
  asm volatile("s_wait_asynccnt 0" ::: "memory");

  // logl = log(alpha[S-1] + alpha[S-2]) + logC - sum_t lse_t
  // s=199 -> lane 28 elem 3, s=200 -> lane 28 elem 4.
  float endv = (lane == 28) ? (a[3] + a[4]) : 0.f;
#pragma unroll
  for (int off = 16; off > 0; off >>= 1) endv += __shfl_xor(endv, off, 32);
  float per = 0.f;
  if (endv > 0.f) {
    const float logl = logf(endv) + (float)logC - sl;
    if (logl > -1e29f) per = -logl * (1.f / (float)L_);   // 'mean' divides by L; zero_infinity
  }
  if (lane == 0) perBatch[b] = per;
}

// ---------------------------------------------------------------------------
// Kernel 3: deterministic final mean over B=64 (no float atomics).
// ---------------------------------------------------------------------------
__global__ void finalize_kernel(const float* __restrict__ perBatch, float* __restrict__ out) {
  if (threadIdx.x == 0 && blockIdx.x == 0) {
    float s = 0.f;
    for (int i = 0; i < B_; ++i) s += perBatch[i];
    out[0] = s * (1.0f / (float)B_);
  }
}

extern "C" void kernel_launch(void* const* d_in, const int* in_sizes, int n_in,
                              void* d_out, int out_size, void* d_ws, size_t ws_size,
                              hipStream_t stream) {
  const float* logits  = (const float*)d_in[0];
  const int*   targets = (const int*)d_in[1];
  float* lse  = (float*)d_ws;                    // 65536 floats
  float* perB = lse + (size_t)B_ * T_;           // 64 floats
  hipLaunchKernelGGL(lse_kernel, dim3((B_ * T_) / 8), dim3(256), 0, stream, logits, lse);
  hipLaunchKernelGGL(ctc_forward_kernel, dim3(B_), dim3(32), 0, stream,
                     logits, targets, lse, perB);
  hipLaunchKernelGGL(finalize_kernel, dim3(1), dim3(1), 0, stream, perB, (float*)d_out);
}